// NetVLADLayer_28535762714832
// MI455X (gfx1250) — compile-verified
//
#include <hip/hip_runtime.h>
#include <hip/hip_bf16.h>
#include <math.h>

// NetVLAD fused pipeline for MI455X (gfx1250, wave32, WMMA f32 16x16x4,
// async global->LDS double-buffered staging). B=16, C=512, N=4096, K=64.

typedef __attribute__((ext_vector_type(2))) float v2f;
typedef __attribute__((ext_vector_type(8))) float v8f;
typedef __attribute__((ext_vector_type(4))) int   v4i;

// Address-space-qualified 16-byte vector types for the async-DMA builtin:
// clang prints AS(1) as __device__ and AS(3) as __shared__ in diagnostics.
typedef __attribute__((address_space(1))) v4i gv4i;   // global
typedef __attribute__((address_space(3))) v4i lv4i;   // LDS

#define BB 16
#define CC 512
#define NN 4096
#define KK 64
#define TILE 64
#define LSTR 76    // LDS row stride (floats): gcd(76,64)=4 (A conflict-free),
                   // 2*76 mod 64 = 24 in [16,48] (B half-wave groups disjoint),
                   // 76*4 = 304 B = 19*16 (b128-aligned rows)
#define EPSV 1e-12f

#if defined(__AMDGCN__) && __has_builtin(__builtin_amdgcn_global_load_async_to_lds_b128)
#define HAS_ASYNC 1
#else
#define HAS_ASYNC 0
#endif

// Copy a 64x64 f32 tile (row stride grs in floats) into padded LDS tile.
// 512 threads, 2 x b128 per thread, per-lane async DMA when available.
__device__ __forceinline__ void stage_tile(const float* __restrict__ g, size_t grs,
                                           float (*__restrict__ l)[LSTR], int tid) {
#if HAS_ASYNC
#pragma unroll
    for (int j = 0; j < 2; ++j) {
        int q = tid + j * 512;
        int row = q >> 4, qc = (q & 15) * 4;
        __builtin_amdgcn_global_load_async_to_lds_b128(
            (gv4i*)(g + (size_t)row * grs + qc),
            (lv4i*)&l[row][qc], 0, 0);
    }
#else
#pragma unroll
    for (int j = 0; j < 2; ++j) {
        int q = tid + j * 512;
        int row = q >> 4, qc = (q & 15) * 4;
        *(float4*)&l[row][qc] = *(const float4*)(g + (size_t)row * grs + qc);
    }
#endif
}

__device__ __forceinline__ void wait_async_le4() {
#if HAS_ASYNC
#if __has_builtin(__builtin_amdgcn_s_wait_asynccnt)
    __builtin_amdgcn_s_wait_asynccnt(4);
#else
    asm volatile("s_wait_asynccnt 0x4" ::: "memory");
#endif
#endif
}

__device__ __forceinline__ void wait_async_le0() {
#if HAS_ASYNC
#if __has_builtin(__builtin_amdgcn_s_wait_asynccnt)
    __builtin_amdgcn_s_wait_asynccnt(0);
#else
    asm volatile("s_wait_asynccnt 0x0" ::: "memory");
#endif
#endif
}

// ---------------------------------------------------------------------------
// Kernel 1: scores = score_w(64x512) @ x_b(512 x 64n-tile), softmax over K,
// store P as (B, N, K). Grid: (N/64, B), block 512 (16 waves).
// ---------------------------------------------------------------------------
__global__ __launch_bounds__(512) void k_scores(const float* __restrict__ x,
                                                const float* __restrict__ sw,
                                                float* __restrict__ P) {
    __shared__ __attribute__((aligned(16))) float lsw[2][TILE][LSTR]; // score_w [k][c]
    __shared__ __attribute__((aligned(16))) float lx [2][TILE][LSTR]; // x       [c][n]

    const int b    = blockIdx.y;
    const int n0   = blockIdx.x * TILE;
    const int tid  = threadIdx.x;
    const int lane = tid & 31;
    const int wave = tid >> 5;          // 0..15
    const int kt   = wave >> 2;         // k-tile 0..3 (M)
    const int nt   = wave & 3;          // n-tile 0..3 (N)
    const int lo16 = lane & 15;
    const int hi2  = (lane >> 4) * 2;   // K-pair select per ISA f32 layout

    const float* xg = x + (size_t)b * CC * NN + n0;   // + c*NN per row

    v8f acc = {};

    // prologue: stage chunk 0 into buffer 0
    stage_tile(sw, CC, lsw[0], tid);
    stage_tile(xg, NN, lx[0], tid);

    const int T = CC / TILE;   // 8 chunks over the C reduction
    for (int t = 0; t < T; ++t) {
        if (t + 1 < T) {   // issue chunk t+1 into other buffer, then wait chunk t
            stage_tile(sw + (t + 1) * TILE, CC, lsw[(t + 1) & 1], tid);
            stage_tile(xg + (size_t)(t + 1) * TILE * NN, NN, lx[(t + 1) & 1], tid);
            wait_async_le4();
        } else {
            wait_async_le0();
        }
        __syncthreads();

        const int u = t & 1;
        const int mrow = kt * 16 + lo16;   // A row (k)
        const int ncol = nt * 16 + lo16;   // B col (n)
        #pragma unroll
        for (int s = 0; s < 16; ++s) {
            const int cb = s * 4 + hi2;
            v2f a = *(const v2f*)&lsw[u][mrow][cb];  // A: score_w (M=k, Kred=c)
            v2f bm;                                   // B: x (Kred=c, N=n), 2addr pair
            bm.x = lx[u][cb][ncol];
            bm.y = lx[u][cb + 1][ncol];
            acc = __builtin_amdgcn_wmma_f32_16x16x4_f32(
                false, a, false, bm, (short)0, acc, false, false);
        }
        __syncthreads();
    }

    // scatter 16x16 C tile into a dead buffer (lx[0] is free after last chunk)
    float (*lS)[LSTR] = lx[0];
    {
        const int half = (lane >> 4) * 8;
        const int n = nt * 16 + lo16;
        #pragma unroll
        for (int r = 0; r < 8; ++r)
            lS[kt * 16 + half + r][n] = acc[r];
    }
    __syncthreads();

    // softmax over K per column n, then write P[b][n0+n][k] (contiguous)
    if (tid < 64) {
        const int n = tid;
        float mx = lS[0][n];
        #pragma unroll 8
        for (int k = 1; k < KK; ++k) mx = fmaxf(mx, lS[k][n]);
        float ssumv = 0.0f;
        #pragma unroll 8
        for (int k = 0; k < KK; ++k) {
            float e = __expf(lS[k][n] - mx);
            ssumv += e;
            lS[k][n] = e;
        }
        const float inv = 1.0f / ssumv;
        float4* dst = (float4*)&P[((size_t)b * NN + n0 + n) * KK];
        #pragma unroll
        for (int q = 0; q < 16; ++q) {
            float4 v;
            v.x = lS[4 * q + 0][n] * inv;
            v.y = lS[4 * q + 1][n] * inv;
            v.z = lS[4 * q + 2][n] * inv;
            v.w = lS[4 * q + 3][n] * inv;
            dst[q] = v;
        }
    }
}

// ---------------------------------------------------------------------------
// Kernel 2: ssum[b][k] = sum_n P[b][n][k]. Grid B, block 256. Deterministic.
// ---------------------------------------------------------------------------
__global__ __launch_bounds__(256) void k_ssum(const float* __restrict__ P,
                                              float* __restrict__ ssum) {
    __shared__ float red[4][KK];
    const int b = blockIdx.x, tid = threadIdx.x;
    const int k = tid & 63, g = tid >> 6;
    float s = 0.0f;
    for (int n = g; n < NN; n += 4)
        s += P[((size_t)b * NN + n) * KK + k];
    red[g][k] = s;
    __syncthreads();
    if (g == 0)
        ssum[b * KK + k] = red[0][k] + red[1][k] + red[2][k] + red[3][k];
}

// ---------------------------------------------------------------------------
// Kernel 3: sx[b] = x_b(C x N) @ P_b^T(N x K). Grid: (C/64, B), block 512.
// ---------------------------------------------------------------------------
__global__ __launch_bounds__(512) void k_sx(const float* __restrict__ x,
                                            const float* __restrict__ P,
                                            float* __restrict__ sx) {
    __shared__ __attribute__((aligned(16))) float lx[2][TILE][LSTR]; // x [c][n]
    __shared__ __attribute__((aligned(16))) float lp[2][TILE][LSTR]; // P [n][k]

    const int b    = blockIdx.y;
    const int c0   = blockIdx.x * TILE;
    const int tid  = threadIdx.x;
    const int lane = tid & 31;
    const int wave = tid >> 5;
    const int ct   = wave >> 2;         // c-tile 0..3 (M)
    const int kt   = wave & 3;          // k-tile 0..3 (N)
    const int lo16 = lane & 15;
    const int hi2  = (lane >> 4) * 2;

    const float* xg = x + (size_t)b * CC * NN + (size_t)c0 * NN;  // rows=c, + n offset
    const float* pg = P + (size_t)b * NN * KK;                    // rows=n, stride KK

    v8f acc = {};

    stage_tile(xg, NN, lx[0], tid);
    stage_tile(pg, KK, lp[0], tid);

    const int T = NN / TILE;   // 64 chunks over the N reduction
    for (int t = 0; t < T; ++t) {
        if (t + 1 < T) {
            stage_tile(xg + (t + 1) * TILE, NN, lx[(t + 1) & 1], tid);
            stage_tile(pg + (size_t)(t + 1) * TILE * KK, KK, lp[(t + 1) & 1], tid);
            wait_async_le4();
        } else {
            wait_async_le0();
        }
        __syncthreads();

        const int u = t & 1;
        const int crow = ct * 16 + lo16;   // A row (c)
        const int kcol = kt * 16 + lo16;   // B col (k)
        #pragma unroll
        for (int s = 0; s < 16; ++s) {
            const int nb = s * 4 + hi2;
            v2f a = *(const v2f*)&lx[u][crow][nb];  // A: x (M=c, Kred=n)
            v2f bm;                                  // B: P^T (Kred=n, N=k), 2addr pair
            bm.x = lp[u][nb][kcol];
            bm.y = lp[u][nb + 1][kcol];
            acc = __builtin_amdgcn_wmma_f32_16x16x4_f32(
                false, a, false, bm, (short)0, acc, false, false);
        }
        __syncthreads();
    }

    // write sx[b][c][k] (coalesced over k within each vgpr row)
    const int half = (lane >> 4) * 8;
    const int k = kt * 16 + lo16;
    #pragma unroll
    for (int r = 0; r < 8; ++r) {
        int c = c0 + ct * 16 + half + r;
        sx[((size_t)b * CC + c) * KK + k] = acc[r];
    }
}

// ---------------------------------------------------------------------------
// Kernel 4: column norms over C; global norm per batch derives from the 64
// column norms (each post-intra-norm column contributes n_k^2/max(n_k,eps)^2).
// scale[b][k] = 1 / (max(n_k,eps) * max(g_b,eps)). Grid B, block 256.
// ---------------------------------------------------------------------------
__global__ __launch_bounds__(256) void k_scale(const float* __restrict__ sx,
                                               const float* __restrict__ centers,
                                               const float* __restrict__ ssum,
                                               float* __restrict__ scale) {
    __shared__ float red[4][KK];
    __shared__ float coln[KK];
    __shared__ float ginv;
    const int b = blockIdx.x, tid = threadIdx.x;
    const int k = tid & 63, g = tid >> 6;
    const float sb = ssum[b * KK + k];
    float acc = 0.0f;
    for (int c = g; c < CC; c += 4) {
        float d = sx[((size_t)b * CC + c) * KK + k] - centers[c * KK + k] * sb;
        acc += d * d;
    }
    red[g][k] = acc;
    __syncthreads();
    if (g == 0)
        coln[k] = sqrtf(red[0][k] + red[1][k] + red[2][k] + red[3][k]);
    __syncthreads();
    if (tid == 0) {
        float gs = 0.0f;
        for (int kk = 0; kk < KK; ++kk) {
            float r = coln[kk];
            float t = r / fmaxf(r, EPSV);
            gs += t * t;
        }
        ginv = 1.0f / fmaxf(sqrtf(gs), EPSV);
    }
    __syncthreads();
    if (g == 0)
        scale[b * KK + k] = ginv / fmaxf(coln[k], EPSV);
}

// ---------------------------------------------------------------------------
// Kernel 5: out[b][c*K+k] = (sx - centers*ssum) * scale. Flat, coalesced.
// ---------------------------------------------------------------------------
__global__ __launch_bounds__(256) void k_out(const float* __restrict__ sx,
                                             const float* __restrict__ centers,
                                             const float* __restrict__ ssum,
                                             const float* __restrict__ scale,
                                             float* __restrict__ out) {
    const size_t i = (size_t)blockIdx.x * 256 + threadIdx.x;   // over B*C*K
    const int k = (int)(i & 63);
    const int c = (int)((i >> 6) & (CC - 1));
    const int b = (int)(i >> 15);
    const float d = sx[i] - centers[c * KK + k] * ssum[b * KK + k];
    out[i] = d * scale[b * KK + k];
}

// ---------------------------------------------------------------------------
extern "C" void kernel_launch(void* const* d_in, const int* in_sizes, int n_in,
                              void* d_out, int out_size, void* d_ws, size_t ws_size,
                              hipStream_t stream) {
    const float* x       = (const float*)d_in[0];   // (B,C,N)
    const float* sw      = (const float*)d_in[1];   // (K,C)
    const float* centers = (const float*)d_in[2];   // (C,K)
    float* out = (float*)d_out;                     // (B, C*K)

    float* P     = (float*)d_ws;                          // (B,N,K) 16 MB
    float* sx    = P    + (size_t)BB * NN * KK;           // (B,C,K)  2 MB
    float* ssum  = sx   + (size_t)BB * CC * KK;           // (B,K)
    float* scale = ssum + (size_t)BB * KK;                // (B,K)

    k_scores<<<dim3(NN / TILE, BB), 512, 0, stream>>>(x, sw, P);
    k_ssum  <<<BB, 256, 0, stream>>>(P, ssum);
    k_sx    <<<dim3(CC / TILE, BB), 512, 0, stream>>>(x, P, sx);
    k_scale <<<BB, 256, 0, stream>>>(sx, centers, ssum, scale);
    k_out   <<<(BB * CC * KK) / 256, 256, 0, stream>>>(sx, centers, ssum, scale, out);
}